// CLinear_6768868459230
// MI455X (gfx1250) — compile-verified
//
#include <hip/hip_runtime.h>

typedef float v2f __attribute__((ext_vector_type(2)));
typedef float v8f __attribute__((ext_vector_type(8)));

#define NCHAN 64
#define FIN_  8
#define FOUT_ 8
#define CDIM_ 128
#define BATCH 256
#define IDIM  512
#define ODIM  512
#define WDSTRIDE 262144               // floats per Wd row (O*I)
#define RING  12                      // B-fragment prefetch depth (24 loads)

// ---------------------------------------------------------------------------
// Kernel 1: fused masked GEMM (wdir = c@Wd + bd) + per-row reductions.
// One wave: (channel, 32-batch tile, 4 of 8 output rows). Two 16x16
// accumulator tiles share every B-fragment load. All global accesses are
// uniform-SGPR-base + one invariant 32-bit lane offset (+imm), so the hot
// loop is saddr-form loads + WMMAs with scalar address stepping.
// ---------------------------------------------------------------------------
__global__ __launch_bounds__(64, 1) void clinear_main(
    const float* __restrict__ x,    // (B, I, 2): [.,.,0]=xv, [.,.,1]=xl
    const float* __restrict__ c,    // (B, CDIM)
    const float* __restrict__ Wd,   // (CDIM, O*I)
    const float* __restrict__ bd,   // (O*I)
    float* __restrict__ s1g,        // (B, O) : sum_i w*xv   (unnormalized w)
    float* __restrict__ s2g,        // (B, O) : sum_i w*w
    float* __restrict__ lseg)       // (B, O) : logsumexp over diag of wdir+xl
{
    const int ch    = blockIdx.x;                  // 0..63
    const int ohalf = threadIdx.x >> 5;            // 0..1 (wave in block)
    const int btile = blockIdx.y;                  // 0..7 (32 batches each)
    const int lane  = threadIdx.x & 31;
    const int lo    = lane & 15;                   // N index within tile
    const int hi    = lane >> 4;                   // lane-half
    const int bbase = btile * 32;

    // Divergent (per-lane) 32-bit element offsets — loop-invariant.
    const int wlane = lo + 2 * hi * WDSTRIDE;            // B-frag lane offset
    const int xlane = (hi * 8) * IDIM * 2 + lo * 2;      // x-tile lane offset
    const int clane = lo * CDIM_ + 2 * hi;               // c-tile lane offset

    // ---- Preload A fragments: c tiles, 2 x (16 batches x 128 k).
    // Layout (16x4 f32 A): lanes 0-15 -> M=lane, K={0,1}; lanes 16-31 -> K={2,3}.
    v2f afrag0[32], afrag1[32];
    {
        const float* cu = c + (size_t)bbase * CDIM_;     // uniform
        #pragma unroll
        for (int kk = 0; kk < 32; ++kk) {
            afrag0[kk] = *(const v2f*)(cu + clane + kk * 4);
            afrag1[kk] = *(const v2f*)(cu + clane + 16 * CDIM_ + kk * 4);
        }
    }

    const int ncols   = (ch + 1) * FIN_;           // needed i-prefix length
    const int nchunks = (ncols + 15) >> 4;         // 16-wide i-chunks
    const int dlo     = ch * FIN_;                 // diag block = [dlo, dlo+8)
    const float NEGINF = -__builtin_inff();

    for (int oo = ohalf * 4; oo < ohalf * 4 + 4; ++oo) {
        const int o = ch * FOUT_ + oo;
        v8f s1a = {}, s2a = {}, lsea = {};
        v8f s1b = {}, s2b = {}, lseb = {};

        for (int ci = 0; ci < nchunks; ++ci) {
            const int i = ci * 16 + lo;                // column this lane owns
            const float* ub  = Wd + (size_t)o * IDIM + (size_t)ci * 16; // uniform
            const float* xub = x + ((size_t)bbase * IDIM + (size_t)ci * 16) * 2;

            // ---- two 16(b) x 16(i) wdir tiles; B fragments pumped through a
            // 12-deep prefetch ring (24 outstanding b32 loads).
            v2f bf[RING];
            #pragma unroll
            for (int p = 0; p < RING; ++p) {
                const float* bp = ub + (size_t)(p * 4) * WDSTRIDE;   // uniform
                bf[p].x = bp[wlane];
                bf[p].y = bp[wlane + WDSTRIDE];
            }

            v8f acc0 = {}, acc1 = {};
            #pragma unroll
            for (int kk = 0; kk < 32; ++kk) {
                const v2f cur = bf[kk % RING];
                if (kk + RING < 32) {
                    const float* bp = ub + (size_t)((kk + RING) * 4) * WDSTRIDE;
                    v2f nb;
                    nb.x = bp[wlane];
                    nb.y = bp[wlane + WDSTRIDE];
                    bf[kk % RING] = nb;
                }
                acc0 = __builtin_amdgcn_wmma_f32_16x16x4_f32(
                        false, afrag0[kk], false, cur, (short)0, acc0, false, false);
                acc1 = __builtin_amdgcn_wmma_f32_16x16x4_f32(
                        false, afrag1[kk], false, cur, (short)0, acc1, false, false);
            }

            const float bdv  = (bd + (size_t)o * IDIM + (size_t)ci * 16)[lo];
            const bool  isz  = (i >= ncols);
            const bool  isd  = (i >= dlo) && (i < dlo + FIN_);
            const bool  last = (ci == nchunks - 1);    // diag block always here

            #pragma unroll
            for (int r = 0; r < 8; ++r) {
                float wd0 = acc0[r] + bdv;                        // wdir[b,o,i]
                float wd1 = acc1[r] + bdv;
                float w0  = isz ? 0.0f : (isd ? __expf(wd0) : wd0);
                float w1  = isz ? 0.0f : (isd ? __expf(wd1) : wd1);
                v2f xx0 = *(const v2f*)(xub + xlane + r * IDIM * 2);        // (xv,xl)
                v2f xx1 = *(const v2f*)(xub + xlane + (16 + r) * IDIM * 2);
                s1a[r] += w0 * xx0.x;  s2a[r] += w0 * w0;
                s1b[r] += w1 * xx1.x;  s2b[r] += w1 * w1;
                if (last) {
                    lsea[r] = isd ? (wd0 + xx0.y) : NEGINF;       // raw t values
                    lseb[r] = isd ? (wd1 + xx1.y) : NEGINF;
                }
            }

            if (last) {
                // stable logsumexp over the 8 diag lanes (within 16-lane half)
                #pragma unroll
                for (int r = 0; r < 8; ++r) {
                    float t0 = lsea[r], t1 = lseb[r];
                    float m0 = t0, m1 = t1;
                    #pragma unroll
                    for (int mk = 8; mk >= 1; mk >>= 1) {
                        m0 = fmaxf(m0, __shfl_xor(m0, mk, 32));
                        m1 = fmaxf(m1, __shfl_xor(m1, mk, 32));
                    }
                    float e0 = isd ? __expf(t0 - m0) : 0.0f;
                    float e1 = isd ? __expf(t1 - m1) : 0.0f;
                    #pragma unroll
                    for (int mk = 8; mk >= 1; mk >>= 1) {
                        e0 += __shfl_xor(e0, mk, 32);
                        e1 += __shfl_xor(e1, mk, 32);
                    }
                    lsea[r] = m0 + __logf(e0);
                    lseb[r] = m1 + __logf(e1);
                }
            }
        }

        // ---- reduce s1,s2 across the 16 N-lanes of each half
        #pragma unroll
        for (int r = 0; r < 8; ++r) {
            float v1 = s1a[r], v2 = s2a[r], v3 = s1b[r], v4 = s2b[r];
            #pragma unroll
            for (int mk = 8; mk >= 1; mk >>= 1) {
                v1 += __shfl_xor(v1, mk, 32);
                v2 += __shfl_xor(v2, mk, 32);
                v3 += __shfl_xor(v3, mk, 32);
                v4 += __shfl_xor(v4, mk, 32);
            }
            s1a[r] = v1; s2a[r] = v2; s1b[r] = v3; s2b[r] = v4;
        }

        if (lo == 0) {   // lanes 0 and 16 own complete sums for 8 batches each
            #pragma unroll
            for (int r = 0; r < 8; ++r) {
                const int b0   = bbase + hi * 8 + r;    // C layout: M = r + 8*hi
                const int idx0 = b0 * ODIM + o;
                s1g[idx0]  = s1a[r];
                s2g[idx0]  = s2a[r];
                lseg[idx0] = lsea[r];
                const int idx1 = (b0 + 16) * ODIM + o;
                s1g[idx1]  = s1b[r];
                s2g[idx1]  = s2b[r];
                lseg[idx1] = lseb[r];
            }
        }
    }
}

// ---------------------------------------------------------------------------
// Kernel 2: wamp = c@Wa+ba, bias = c@Wb+bb (tiny dots), then finalize:
//   y      = s1 / sqrt(s2) * exp(wamp) + bias
//   logdet = lse - 0.5*log(s2) + wamp
// ---------------------------------------------------------------------------
__global__ __launch_bounds__(256) void clinear_finalize(
    const float* __restrict__ c,
    const float* __restrict__ Wa, const float* __restrict__ ba,
    const float* __restrict__ Wb, const float* __restrict__ bb,
    const float* __restrict__ s1g, const float* __restrict__ s2g,
    const float* __restrict__ lseg,
    float* __restrict__ out)        // (B, O, 2)
{
    __shared__ float csh[CDIM_];
    const int b = blockIdx.x;
    const int o = blockIdx.y * 256 + threadIdx.x;
    if (threadIdx.x < CDIM_)
        csh[threadIdx.x] = c[(size_t)b * CDIM_ + threadIdx.x];
    __syncthreads();

    float wamp = ba[o];
    float bias = bb[o];
    #pragma unroll 4
    for (int k = 0; k < CDIM_; ++k) {
        const float cv = csh[k];
        wamp += cv * Wa[(size_t)k * ODIM + o];
        bias += cv * Wb[(size_t)k * ODIM + o];
    }

    const int idx = b * ODIM + o;
    const float v1 = s1g[idx];
    const float v2 = s2g[idx];
    const float l  = lseg[idx];

    out[idx * 2 + 0] = v1 * rsqrtf(v2) * __expf(wamp) + bias;
    out[idx * 2 + 1] = l - 0.5f * __logf(v2) + wamp;
}

// ---------------------------------------------------------------------------
extern "C" void kernel_launch(void* const* d_in, const int* in_sizes, int n_in,
                              void* d_out, int out_size, void* d_ws, size_t ws_size,
                              hipStream_t stream)
{
    const float* x  = (const float*)d_in[0];
    const float* c  = (const float*)d_in[1];
    const float* Wd = (const float*)d_in[2];
    const float* bd = (const float*)d_in[3];
    const float* Wa = (const float*)d_in[4];
    const float* ba = (const float*)d_in[5];
    const float* Wb = (const float*)d_in[6];
    const float* bb = (const float*)d_in[7];
    float* out = (float*)d_out;

    float* s1  = (float*)d_ws;                 // B*O floats
    float* s2  = s1 + BATCH * ODIM;            // B*O floats
    float* lse = s2 + BATCH * ODIM;            // B*O floats (1.5 MB total)

    dim3 g1(NCHAN, 8), b1(64);                 // 64 ch x 8 btiles, 2 waves/block
    clinear_main<<<g1, b1, 0, stream>>>(x, c, Wd, bd, s1, s2, lse);

    dim3 g2(BATCH, 2), b2(256);
    clinear_finalize<<<g2, b2, 0, stream>>>(c, Wa, ba, Wb, bb, s1, s2, lse, out);
}